// MTCNN_26345329393838
// MI455X (gfx1250) — compile-verified
//
#include <hip/hip_runtime.h>
#include <hip/hip_bf16.h>

typedef float v2f __attribute__((ext_vector_type(2)));
typedef float v8f __attribute__((ext_vector_type(8)));

#define WAVES_PER_BLOCK 4

// ---------------------------------------------------------------------------
// Implicit-GEMM convolution using fp32 WMMA 16x16x4.
//   GEMM: M = Co, N = Nbatch*Ho*Wo, K = Ci*KH*KW (taps padded to multiple of 4)
//   A[m][k] = weight (OIHW) staged in LDS (rows >= Co zero-filled)
//   B[k][n] = on-the-fly im2col gather (addresses always clamped valid;
//             garbage only lands in discarded D rows/cols; K-pad zeroed in A)
// Epilogue fuses bias + PReLU. One wave per 16x16 output tile; the 4 waves of
// a block share one 16-row weight tile in LDS.
// ---------------------------------------------------------------------------
template <int KH, int KW>
__global__ __launch_bounds__(128) void conv_prelu_wmma(
    const float* __restrict__ in, const float* __restrict__ wgt,
    const float* __restrict__ bias, const float* __restrict__ alpha,
    float* __restrict__ out,
    int N, int Ci, int H, int W, int Co, int Ho, int Wo)
{
  extern __shared__ float ldsW[];              // [16][K]
  constexpr int TAPS = KH * KW;
  constexpr int NG   = (TAPS + 3) / 4;         // 4-wide K groups per ci
  const int K = Ci * TAPS;

  // ---- cooperative stage of the 16 x K weight tile into LDS ----
  const int tileM = blockIdx.y;
  for (int idx = threadIdx.x; idx < 16 * K; idx += 128) {
    int row = idx / K, col = idx - row * K;    // outside the hot loop
    int m = tileM * 16 + row;
    ldsW[idx] = (m < Co) ? wgt[(long long)m * K + col] : 0.f;
  }
  __syncthreads();

  const long long P = (long long)N * Ho * Wo;
  const int lane = threadIdx.x & 31;
  const int wave = threadIdx.x >> 5;
  const long long tileN = (long long)blockIdx.x * WAVES_PER_BLOCK + wave;
  if (tileN >= ((P + 15) >> 4)) return;        // wave-uniform, after barrier

  const int hi  = lane >> 4;                   // K-pair select within group
  const int sub = lane & 15;
  const long long n = tileN * 16 + sub;        // B col (flat spatial)
  const bool nval = (n < P);
  const long long nc = nval ? n : 0;           // clamp -> always-valid address
  const int wo = (int)(nc % Wo);
  const long long t1 = nc / Wo;
  const int ho = (int)(t1 % Ho);
  const int nb = (int)(t1 / Ho);
  const long long HW = (long long)H * W;
  const float* bIn  = in + ((long long)nb * Ci) * HW + (long long)ho * W + wo;
  const float* aRow = ldsW + sub * K;          // this lane's A row

  // Precompute tap offsets for each group (constexpr KW -> cheap)
  int  tIdx[NG * 2];
  int  tOff[NG * 2];
  bool aOk [NG * 2];
#pragma unroll
  for (int g = 0; g < NG; ++g) {
#pragma unroll
    for (int j = 0; j < 2; ++j) {
      int t = g * 4 + hi * 2 + j;
      bool v = (t < TAPS);                     // false only in padded slots
      int tt = v ? t : 0;
      tIdx[g * 2 + j] = tt;
      tOff[g * 2 + j] = (tt / KW) * W + (tt % KW);
      aOk [g * 2 + j] = v;
    }
  }

  v8f acc = {0.f, 0.f, 0.f, 0.f, 0.f, 0.f, 0.f, 0.f};
  for (int ci = 0; ci < Ci; ++ci) {
    const float* ar = aRow + ci * TAPS;
#pragma unroll
    for (int g = 0; g < NG; ++g) {
      v2f av, bv;
      float a0 = ar[tIdx[g * 2 + 0]];
      float a1 = ar[tIdx[g * 2 + 1]];
      av.x = aOk[g * 2 + 0] ? a0 : 0.f;        // zero K-padding (A side only)
      av.y = aOk[g * 2 + 1] ? a1 : 0.f;
      bv.x = bIn[tOff[g * 2 + 0]];             // unconditional, clamped-safe
      bv.y = bIn[tOff[g * 2 + 1]];
      acc = __builtin_amdgcn_wmma_f32_16x16x4_f32(
          false, av, false, bv, (short)0, acc, false, false);
    }
    bIn += HW;
  }

  if (!nval) return;                           // divergence only after WMMAs
  const long long HoWo = (long long)Ho * Wo;
  const long long outBase = (((long long)nb * Co) * Ho + ho) * Wo + wo;
#pragma unroll
  for (int v = 0; v < 8; ++v) {
    int co = tileM * 16 + v + hi * 8;          // D: m = v + 8*(lane/16)
    if (co < Co) {
      float x = acc[v] + bias[co];
      x = (x >= 0.f) ? x : alpha[co] * x;      // PReLU
      out[outBase + (long long)co * HoWo] = x;
    }
  }
}

// ---------------------------------------------------------------------------
// Fully-connected layer with fused (0,3,2,1)-permute gather + bias + PReLU.
// K iterated as (w2, h, c-blocks of 4) which matches flatten order exactly:
//   f = (w2*H + h)*C + c  -> no div/mod anywhere in the loop.
// ---------------------------------------------------------------------------
__global__ __launch_bounds__(128) void linear_prelu_wmma(
    const float* __restrict__ x, const float* __restrict__ wgt,
    const float* __restrict__ bias, const float* __restrict__ alpha,
    float* __restrict__ out,
    int Nb, int Fin, int Fout, int C, int H, int W)
{
  const int lane = threadIdx.x & 31;
  const int wave = threadIdx.x >> 5;
  const int tileN = blockIdx.x * WAVES_PER_BLOCK + wave;
  if (tileN >= ((Nb + 15) >> 4)) return;
  const int tileM = blockIdx.y;
  const int hi = lane >> 4, sub = lane & 15;
  const int m = tileM * 16 + sub;
  const int n = tileN * 16 + sub;
  const bool nval = (n < Nb);
  const int mi = (m < Fout) ? m : 0;           // clamp: bad rows -> discarded D
  const int nn = nval ? n : 0;                 // clamp: bad cols -> discarded D
  const long long HWd = (long long)H * W;
  const float* xn = x + (long long)nn * C * HWd;
  const float* wm = wgt + (long long)mi * Fin;

  v8f acc = {0.f, 0.f, 0.f, 0.f, 0.f, 0.f, 0.f, 0.f};
  for (int w2 = 0; w2 < W; ++w2) {
    for (int h = 0; h < H; ++h) {
      const float* bp = xn + (long long)h * W + w2;
      const int kb = (w2 * H + h) * C;
      for (int c0 = 0; c0 < C; c0 += 4) {      // C is a multiple of 4
        const int ca = c0 + hi * 2;
        v2f av, bv;
        av.x = wm[kb + ca];
        av.y = wm[kb + ca + 1];
        bv.x = bp[(long long)ca * HWd];
        bv.y = bp[(long long)(ca + 1) * HWd];
        acc = __builtin_amdgcn_wmma_f32_16x16x4_f32(
            false, av, false, bv, (short)0, acc, false, false);
      }
    }
  }

  if (!nval) return;
#pragma unroll
  for (int v = 0; v < 8; ++v) {
    int j = tileM * 16 + v + hi * 8;
    if (j < Fout) {
      float y = acc[v] + bias[j];
      y = (y >= 0.f) ? y : alpha[j] * y;
      out[(long long)n * Fout + j] = y;
    }
  }
}

// ---------------------------------------------------------------------------
// ceil-mode max pool (right/bottom implicit -inf padding); 1 thread / output
// ---------------------------------------------------------------------------
__global__ __launch_bounds__(256) void maxpool_ceil_kernel(
    const float* __restrict__ in, float* __restrict__ out,
    int N, int C, int H, int W, int k, int s, int Ho, int Wo)
{
  long long i = (long long)blockIdx.x * blockDim.x + threadIdx.x;
  long long tot = (long long)N * C * Ho * Wo;
  if (i >= tot) return;
  int wo = (int)(i % Wo); long long t = i / Wo;
  int ho = (int)(t % Ho); t /= Ho;
  int c = (int)(t % C); int nb = (int)(t / C);
  const float* p = in + (((long long)nb * C + c) * H) * W;
  float mx = -3.402823466e38f;
  const int h0 = ho * s, w0 = wo * s;
  for (int dh = 0; dh < k; ++dh) {
    int h = h0 + dh; if (h >= H) break;
    for (int dw = 0; dw < k; ++dw) {
      int w = w0 + dw; if (w >= W) break;
      float v = p[(long long)h * W + w];
      mx = v > mx ? v : mx;
    }
  }
  out[i] = mx;
}

// ---------------------------------------------------------------------------
// P-Net head: fused 1x1 cls conv + softmax(axis=1) and 1x1 box conv.
// ---------------------------------------------------------------------------
__global__ __launch_bounds__(256) void pnet_head_kernel(
    const float* __restrict__ x,
    const float* __restrict__ clsW, const float* __restrict__ clsB,
    const float* __restrict__ boxW, const float* __restrict__ boxB,
    float* __restrict__ pb, float* __restrict__ pa,
    int N, int Cin, long long P)
{
  long long i = (long long)blockIdx.x * blockDim.x + threadIdx.x;
  if (i >= (long long)N * P) return;
  long long pix = i % P; int nb = (int)(i / P);
  const float* xp = x + ((long long)nb * Cin) * P + pix;
  float c0 = clsB[0], c1 = clsB[1];
  float b0 = boxB[0], b1 = boxB[1], b2 = boxB[2], b3 = boxB[3];
  for (int c = 0; c < Cin; ++c) {
    float v = xp[(long long)c * P];
    c0 += v * clsW[c];            c1 += v * clsW[Cin + c];
    b0 += v * boxW[c];            b1 += v * boxW[Cin + c];
    b2 += v * boxW[2 * Cin + c];  b3 += v * boxW[3 * Cin + c];
  }
  float mx = fmaxf(c0, c1);
  float e0 = __expf(c0 - mx), e1 = __expf(c1 - mx);
  float inv = 1.f / (e0 + e1);
  pa[((long long)nb * 2 + 0) * P + pix] = e0 * inv;
  pa[((long long)nb * 2 + 1) * P + pix] = e1 * inv;
  pb[((long long)nb * 4 + 0) * P + pix] = b0;
  pb[((long long)nb * 4 + 1) * P + pix] = b1;
  pb[((long long)nb * 4 + 2) * P + pix] = b2;
  pb[((long long)nb * 4 + 3) * P + pix] = b3;
}

// ---------------------------------------------------------------------------
// R/O-Net head: cls(2)+softmax, box(4), optional landmarks. 1 thread / sample
// ---------------------------------------------------------------------------
__global__ __launch_bounds__(256) void fc_head_kernel(
    const float* __restrict__ x, int Nb, int Fin,
    const float* __restrict__ clsW, const float* __restrict__ clsB,
    const float* __restrict__ boxW, const float* __restrict__ boxB,
    const float* __restrict__ lmkW, const float* __restrict__ lmkB, int nLmk,
    float* __restrict__ outBox, float* __restrict__ outLmk,
    float* __restrict__ outCls)
{
  int n = blockIdx.x * blockDim.x + threadIdx.x;
  if (n >= Nb) return;
  const float* xr = x + (long long)n * Fin;
  float c0 = clsB[0], c1 = clsB[1];
  float bb0 = boxB[0], bb1 = boxB[1], bb2 = boxB[2], bb3 = boxB[3];
  for (int k = 0; k < Fin; ++k) {
    float v = xr[k];
    c0  += v * clsW[k];           c1  += v * clsW[Fin + k];
    bb0 += v * boxW[k];           bb1 += v * boxW[Fin + k];
    bb2 += v * boxW[2 * Fin + k]; bb3 += v * boxW[3 * Fin + k];
  }
  float mx = fmaxf(c0, c1);
  float e0 = __expf(c0 - mx), e1 = __expf(c1 - mx);
  float inv = 1.f / (e0 + e1);
  outCls[(long long)n * 2 + 0] = e0 * inv;
  outCls[(long long)n * 2 + 1] = e1 * inv;
  outBox[(long long)n * 4 + 0] = bb0;
  outBox[(long long)n * 4 + 1] = bb1;
  outBox[(long long)n * 4 + 2] = bb2;
  outBox[(long long)n * 4 + 3] = bb3;
  for (int j = 0; j < nLmk; ++j) {
    float a = lmkB[j];
    for (int k = 0; k < Fin; ++k) a += xr[k] * lmkW[(long long)j * Fin + k];
    outLmk[(long long)n * nLmk + j] = a;
  }
}

// ---------------------------------------------------------------------------

static inline int ceil_pool(int H, int k, int s) { return (H - k + s - 1) / s + 1; }

extern "C" void kernel_launch(void* const* d_in, const int* in_sizes, int n_in,
                              void* d_out, int out_size, void* d_ws, size_t ws_size,
                              hipStream_t stream) {
  (void)in_sizes; (void)n_in; (void)out_size; (void)ws_size;
  const float* img     = (const float*)d_in[0];
  const float* crops24 = (const float*)d_in[1];
  const float* crops48 = (const float*)d_in[2];
  // pnet params (dict order)
  const float *p_c1w=(const float*)d_in[3],  *p_c1b=(const float*)d_in[4],  *p_a1=(const float*)d_in[5];
  const float *p_c2w=(const float*)d_in[6],  *p_c2b=(const float*)d_in[7],  *p_a2=(const float*)d_in[8];
  const float *p_c3w=(const float*)d_in[9],  *p_c3b=(const float*)d_in[10], *p_a3=(const float*)d_in[11];
  const float *p_clsw=(const float*)d_in[12],*p_clsb=(const float*)d_in[13];
  const float *p_boxw=(const float*)d_in[14],*p_boxb=(const float*)d_in[15];
  // rnet params
  const float *r_c1w=(const float*)d_in[16], *r_c1b=(const float*)d_in[17], *r_a1=(const float*)d_in[18];
  const float *r_c2w=(const float*)d_in[19], *r_c2b=(const float*)d_in[20], *r_a2=(const float*)d_in[21];
  const float *r_c3w=(const float*)d_in[22], *r_c3b=(const float*)d_in[23], *r_a3=(const float*)d_in[24];
  const float *r_fcw=(const float*)d_in[25], *r_fcb=(const float*)d_in[26], *r_fca=(const float*)d_in[27];
  const float *r_clsw=(const float*)d_in[28],*r_clsb=(const float*)d_in[29];
  const float *r_boxw=(const float*)d_in[30],*r_boxb=(const float*)d_in[31];
  // onet params
  const float *o_c1w=(const float*)d_in[32], *o_c1b=(const float*)d_in[33], *o_a1=(const float*)d_in[34];
  const float *o_c2w=(const float*)d_in[35], *o_c2b=(const float*)d_in[36], *o_a2=(const float*)d_in[37];
  const float *o_c3w=(const float*)d_in[38], *o_c3b=(const float*)d_in[39], *o_a3=(const float*)d_in[40];
  const float *o_c4w=(const float*)d_in[41], *o_c4b=(const float*)d_in[42], *o_a4=(const float*)d_in[43];
  const float *o_fcw=(const float*)d_in[44], *o_fcb=(const float*)d_in[45], *o_fca=(const float*)d_in[46];
  const float *o_clsw=(const float*)d_in[47],*o_clsb=(const float*)d_in[48];
  const float *o_boxw=(const float*)d_in[49],*o_boxb=(const float*)d_in[50];
  const float *o_lmkw=(const float*)d_in[51],*o_lmkb=(const float*)d_in[52];

  // Output tuple layout: (pb, pa, rb, ra, ob, oc, oa), flat fp32
  float* out = (float*)d_out;
  float* pb = out;                 // 8*4*355*635
  float* pa = pb + 7213600LL;      // 8*2*355*635
  float* rb = pa + 3606800LL;      // 4096*4
  float* ra = rb + 16384LL;        // 4096*2
  float* ob = ra + 8192LL;         // 1024*4
  float* oc = ob + 4096LL;         // 1024*10
  float* oa = oc + 10240LL;        // 1024*2

  // Two ping-pong workspace regions; A is 73,407,360 floats, 960 short of p_x1 (73,408,320),
  // B sized for p_x3 (57,708,800 floats). Every other intermediate fits.
  float* bufA = (float*)d_ws;
  float* bufB = (float*)((char*)d_ws + 293629440ull);

  auto conv = [&](const float* in, const float* w, const float* b, const float* a,
                  float* o, int N, int Ci, int H, int W, int Co, int Kh, int Kw) {
    int Ho = H - Kh + 1, Wo = W - Kw + 1;
    long long P = (long long)N * Ho * Wo;
    long long tilesN = (P + 15) >> 4;
    dim3 grid((unsigned)((tilesN + WAVES_PER_BLOCK - 1) / WAVES_PER_BLOCK),
              (unsigned)((Co + 15) / 16));
    size_t smem = (size_t)16 * Ci * Kh * Kw * sizeof(float);
    if (Kh == 3)
      conv_prelu_wmma<3, 3><<<grid, dim3(128), smem, stream>>>(
          in, w, b, a, o, N, Ci, H, W, Co, Ho, Wo);
    else
      conv_prelu_wmma<2, 2><<<grid, dim3(128), smem, stream>>>(
          in, w, b, a, o, N, Ci, H, W, Co, Ho, Wo);
  };
  auto pool = [&](const float* in, float* o, int N, int C, int H, int W, int k, int s) {
    int Ho = ceil_pool(H, k, s), Wo = ceil_pool(W, k, s);
    long long tot = (long long)N * C * Ho * Wo;
    maxpool_ceil_kernel<<<(unsigned)((tot + 255) / 256), 256, 0, stream>>>(
        in, o, N, C, H, W, k, s, Ho, Wo);
  };
  auto fc = [&](const float* x, const float* w, const float* b, const float* a,
                float* o, int Nb, int Fin, int Fout, int C, int H, int W) {
    dim3 grid((unsigned)(((Nb + 15) / 16 + WAVES_PER_BLOCK - 1) / WAVES_PER_BLOCK),
              (unsigned)((Fout + 15) / 16));
    linear_prelu_wmma<<<grid, dim3(128), 0, stream>>>(x, w, b, a, o,
        Nb, Fin, Fout, C, H, W);
  };

  // ---------------- P-Net: (8,3,720,1280) ----------------
  conv(img,  p_c1w, p_c1b, p_a1, bufA, 8, 3, 720, 1280, 10, 3, 3);   // -> 718x1278
  pool(bufA, bufB, 8, 10, 718, 1278, 2, 2);                          // -> 359x639
  conv(bufB, p_c2w, p_c2b, p_a2, bufA, 8, 10, 359, 639, 16, 3, 3);   // -> 357x637
  conv(bufA, p_c3w, p_c3b, p_a3, bufB, 8, 16, 357, 637, 32, 3, 3);   // -> 355x635
  {
    long long P = 355LL * 635LL;
    long long tot = 8LL * P;
    pnet_head_kernel<<<(unsigned)((tot + 255) / 256), 256, 0, stream>>>(
        bufB, p_clsw, p_clsb, p_boxw, p_boxb, pb, pa, 8, 32, P);
  }

  // ---------------- R-Net: (4096,3,24,24) ----------------
  conv(crops24, r_c1w, r_c1b, r_a1, bufA, 4096, 3, 24, 24, 28, 3, 3); // -> 22x22
  pool(bufA, bufB, 4096, 28, 22, 22, 3, 2);                           // -> 11x11
  conv(bufB, r_c2w, r_c2b, r_a2, bufA, 4096, 28, 11, 11, 48, 3, 3);   // -> 9x9
  pool(bufA, bufB, 4096, 48, 9, 9, 3, 2);                             // -> 4x4
  conv(bufB, r_c3w, r_c3b, r_a3, bufA, 4096, 48, 4, 4, 64, 2, 2);     // -> 3x3
  fc(bufA, r_fcw, r_fcb, r_fca, bufB, 4096, 576, 128, 64, 3, 3);      // permute-fused
  fc_head_kernel<<<(4096 + 255) / 256, 256, 0, stream>>>(
      bufB, 4096, 128, r_clsw, r_clsb, r_boxw, r_boxb,
      (const float*)nullptr, (const float*)nullptr, 0, rb, (float*)nullptr, ra);

  // ---------------- O-Net: (1024,3,48,48) ----------------
  conv(crops48, o_c1w, o_c1b, o_a1, bufA, 1024, 3, 48, 48, 32, 3, 3); // -> 46x46
  pool(bufA, bufB, 1024, 32, 46, 46, 3, 2);                           // -> 23x23
  conv(bufB, o_c2w, o_c2b, o_a2, bufA, 1024, 32, 23, 23, 64, 3, 3);   // -> 21x21
  pool(bufA, bufB, 1024, 64, 21, 21, 3, 2);                           // -> 10x10
  conv(bufB, o_c3w, o_c3b, o_a3, bufA, 1024, 64, 10, 10, 64, 3, 3);   // -> 8x8
  pool(bufA, bufB, 1024, 64, 8, 8, 2, 2);                             // -> 4x4
  conv(bufB, o_c4w, o_c4b, o_a4, bufA, 1024, 64, 4, 4, 128, 2, 2);    // -> 3x3
  fc(bufA, o_fcw, o_fcb, o_fca, bufB, 1024, 1152, 256, 128, 3, 3);    // permute-fused
  fc_head_kernel<<<(1024 + 255) / 256, 256, 0, stream>>>(
      bufB, 1024, 256, o_clsw, o_clsb, o_boxw, o_boxb,
      o_lmkw, o_lmkb, 10, ob, oc, oa);
}